// DeformableAttention_25168508354795
// MI455X (gfx1250) — compile-verified
//
#include <hip/hip_runtime.h>
#include <hip/hip_bf16.h>

#if __has_include(<hip/amd_detail/amd_gfx1250_TDM.h>)
#define CDNA5_TDM_6ARG 1
#else
#define CDNA5_TDM_6ARG 0
#endif

// ---------------------------------------------------------------------------
// Multi-scale deformable attention for MI455X (gfx1250, wave32, WMMA + TDM).
//   1. V   = X @ Wv + bv          (WMMA f32 GEMM, A streamed from feats)
//   2. Q   = X @ Wq + bq          (WMMA f32 GEMM, A streamed from feats)
//   3. OFF = Q @ Woff + boff      (WMMA f32 GEMM, A tile via TDM->LDS)
//   4. ATT = softmax16(Q @ Wattn + battn)
//   5. AGG = bilinear-sample(V, OFF) weighted by ATT
//   6. out = AGG @ Wout + bout, scattered straight into [B,C,H,W,T] layout
// ---------------------------------------------------------------------------

typedef float v2f  __attribute__((ext_vector_type(2)));
typedef float v8f  __attribute__((ext_vector_type(8)));
typedef unsigned int u32x4 __attribute__((ext_vector_type(4)));
typedef int   i32x4 __attribute__((ext_vector_type(4)));
typedef int   i32x8 __attribute__((ext_vector_type(8)));

#define NUM_HEADS  8
#define NUM_LEVELS 4
#define NUM_POINTS 4
#define HEAD_DIM   32
#define EMBED      256
#define FRAMES     2
#define ROWS_TOTAL 21760        // BT * (4096+1024+256+64)
#define ROWT       64           // rows per GEMM block
#define ROW_BLKS   340          // ROWS_TOTAL / 64
#define LSTR       260          // LDS row stride: 256 + 4 pad dwords (bank-conflict-free)

// Row space: [level][bt][n], level row bases = BT * cumulative HW.
__device__ __forceinline__ void row_to_lbn(int row, int& l, int& bt, int& n) {
  if (row < 16384)      { l = 0; int r = row;         bt = r >> 12; n = r & 4095; }
  else if (row < 20480) { l = 1; int r = row - 16384; bt = r >> 10; n = r & 1023; }
  else if (row < 21504) { l = 2; int r = row - 20480; bt = r >> 8;  n = r & 255;  }
  else                  { l = 3; int r = row - 21504; bt = r >> 6;  n = r & 63;   }
}

__device__ __forceinline__ v8f wmma4(v2f a, v2f b, v8f c) {
  return __builtin_amdgcn_wmma_f32_16x16x4_f32(false, a, false, b, (short)0, c,
                                               false, false);
}

// ---------------------------------------------------------------------------
// WMMA GEMM: out[row, col] = sum_k A[row,k] * W[k,col] + bias[col]
//   AMODE 0: A row-major [rows x 256] in workspace -> A tile fetched by the
//            Tensor Data Mover with LDS padding (4 dwords / 256-dword line).
//   AMODE 1: A gathered from feats tensors (innermost stride = T, not TDM-able)
//   OMODE 0: out row-major [rows x NCOL]
//   OMODE 1: scatter to concatenated [B,C,Hl,Wl,T] outputs
// Block 256 thr = 8 waves; wave w -> cols [blockIdx.y*128 + w*16); 4 row-subtiles.
// ---------------------------------------------------------------------------
template<int NCOL, int AMODE, int OMODE>
__global__ __launch_bounds__(256)
void gemm_wmma_kernel(const float* __restrict__ A,
                      const float* __restrict__ f0, const float* __restrict__ f1,
                      const float* __restrict__ f2, const float* __restrict__ f3,
                      const float* __restrict__ W, const float* __restrict__ bias,
                      float* __restrict__ out)
{
  __shared__ float As[ROWT * LSTR];        // padded row-major A tile (~65 KB)

  const int tid     = threadIdx.x;
  const int wave    = tid >> 5;
  const int lane    = tid & 31;
  const int rowBase = blockIdx.x * ROWT;
  const int colBase = blockIdx.y * 128 + wave * 16;

  int l, bt, n0;
  row_to_lbn(rowBase, l, bt, n0);

  // ---- stage 64x256 A tile into LDS (padded layout As[i*LSTR + k]) ----
  if (AMODE == 0) {
#if __has_builtin(__builtin_amdgcn_tensor_load_to_lds)
    if (wave == 0) {   // one wave drives the TDM; scalar op, EXEC-independent
      const float* gsrc = A + (size_t)rowBase * EMBED;
      const unsigned long long ga = (unsigned long long)(uintptr_t)gsrc;
      const unsigned lds = (unsigned)(uintptr_t)(void*)&As[0];
      // D# group0: count=1 | lds_addr | 57-bit global addr | type=2 ("image")
      u32x4 g0 = { 1u, lds, (unsigned)(ga & 0xffffffffu),
                   (unsigned)((ga >> 32) & 0x1ffffffu) | (2u << 30) };
      // D# group1: data_size=4B, pad_enable, pad_interval=256dw, pad_amount=4dw,
      //            tensor_dim0=256, tensor_dim1=64, tile_dim0=256, tile_dim1=64,
      //            tensor_dim0_stride=256
      i32x8 g1 = { (2 << 16) | (1 << 20) | (7 << 22) | (3 << 25),
                   (int)(256u << 16),   // tensor_dim0 low16 -> e1[31:16]
                   (int)(64u  << 16),   // tensor_dim1 low16 -> e2[31:16]
                   (int)(256u << 16),   // tile_dim0         -> e3[31:16]
                   64,                  // tile_dim1         -> e4[15:0]
                   256,                 // tensor_dim0_stride low32
                   0, 0 };
      i32x4 gz = { 0, 0, 0, 0 };
#if CDNA5_TDM_6ARG
      i32x8 gz8 = { 0, 0, 0, 0, 0, 0, 0, 0 };
      __builtin_amdgcn_tensor_load_to_lds(g0, g1, gz, gz, gz8, 0);
#else
      __builtin_amdgcn_tensor_load_to_lds(g0, g1, gz, gz, 0);
#endif
      __builtin_amdgcn_s_wait_tensorcnt(0);
    }
#else
    {
      const int i = tid & 63;
      const int g = tid >> 6;
      #pragma unroll 8
      for (int j = 0; j < 64; ++j) {
        const int c = g * 64 + j;
        As[i * LSTR + c] = A[(size_t)(rowBase + i) * EMBED + c];
      }
    }
#endif
  } else {
    constexpr int HW[4] = {4096, 1024, 256, 64};
    const float* f = (l == 0) ? f0 : (l == 1) ? f1 : (l == 2) ? f2 : f3;
    const int b   = bt >> 1;
    const int tt  = bt & 1;
    const int HWl = HW[l];
    const int i   = tid & 63;              // lanes -> consecutive n (coalesced)
    const int g   = tid >> 6;
    #pragma unroll 8
    for (int j = 0; j < 64; ++j) {
      const int c = g * 64 + j;
      As[i * LSTR + c] = f[(((size_t)(b * EMBED + c) * HWl) + (n0 + i)) * FRAMES + tt];
    }
  }
  __syncthreads();

  // ---- WMMA main loop: one B frag feeds 4 row-subtiles; B is pipelined ----
  // fp32 A frag: lanes 0-15 hold {K=k,k+1}, lanes 16-31 {k+2,k+3}.
  const int li = lane & 15;
  const int ko = (lane >> 4) << 1;         // 0 or 2
  v8f acc0 = {0.f,0.f,0.f,0.f,0.f,0.f,0.f,0.f};
  v8f acc1 = acc0, acc2 = acc0, acc3 = acc0;

  const float* Wp = W + colBase + li;
  __builtin_prefetch(Wp,                      0, 1);   // global_prefetch_b8 burst
  __builtin_prefetch(Wp + (size_t)8  * NCOL,  0, 1);
  __builtin_prefetch(Wp + (size_t)16 * NCOL,  0, 1);
  __builtin_prefetch(Wp + (size_t)24 * NCOL,  0, 1);

  v2f b;
  b.x = Wp[(size_t)ko * NCOL];
  b.y = Wp[(size_t)(ko + 1) * NCOL];

  #pragma unroll 4
  for (int k = 0; k < EMBED; k += 4) {
    const int kn = (k + 4 < EMBED) ? (k + 4) : k;       // pipelined next B frag
    v2f bn;
    bn.x = Wp[(size_t)(kn + ko) * NCOL];
    bn.y = Wp[(size_t)(kn + ko + 1) * NCOL];

    const float* ap = &As[li * LSTR + k + ko];
    v2f a0 = { ap[0 * 16 * LSTR], ap[0 * 16 * LSTR + 1] };
    v2f a1 = { ap[1 * 16 * LSTR], ap[1 * 16 * LSTR + 1] };
    v2f a2 = { ap[2 * 16 * LSTR], ap[2 * 16 * LSTR + 1] };
    v2f a3 = { ap[3 * 16 * LSTR], ap[3 * 16 * LSTR + 1] };

    acc0 = wmma4(a0, b, acc0);             // 4 independent WMMAs per B frag
    acc1 = wmma4(a1, b, acc1);
    acc2 = wmma4(a2, b, acc2);
    acc3 = wmma4(a3, b, acc3);
    b = bn;
  }

  // ---- epilogue: C/D layout lane<16 -> M=r, lane>=16 -> M=8+r, N=lane&15 ----
  const int   mh   = (lane >> 4) << 3;     // 0 or 8
  const int   col  = colBase + li;
  const float bcol = bias[col];
  v8f accs[4] = {acc0, acc1, acc2, acc3};

  if (OMODE == 0) {
    #pragma unroll
    for (int st = 0; st < 4; ++st) {
      float* op = out + (size_t)(rowBase + st * 16 + mh) * NCOL + col;
      #pragma unroll
      for (int r = 0; r < 8; ++r)
        op[(size_t)r * NCOL] = accs[st][r] + bcol;
    }
  } else {
    constexpr int    HW[4]     = {4096, 1024, 256, 64};
    constexpr size_t OSTART[4] = {0, 4194304, 5242880, 5505024};
    const int b   = bt >> 1;
    const int tt  = bt & 1;
    const int HWl = HW[l];
    float* op = out + OSTART[l] + (((size_t)(b * EMBED + col)) * HWl) * FRAMES + tt;
    #pragma unroll
    for (int st = 0; st < 4; ++st)
      #pragma unroll
      for (int r = 0; r < 8; ++r)
        op[(size_t)(n0 + st * 16 + mh + r) * FRAMES] = accs[st][r] + bcol;
  }
}

// ---------------------------------------------------------------------------
// Softmax over the 16 (level,point) logits per (row, head); contiguous 16.
// ---------------------------------------------------------------------------
__global__ void softmax16_kernel(float* __restrict__ att, int total)
{
  const int i = blockIdx.x * blockDim.x + threadIdx.x;
  if (i >= total) return;
  float* p = att + (size_t)i * 16;
  float m = p[0];
  #pragma unroll
  for (int j = 1; j < 16; ++j) m = fmaxf(m, p[j]);
  float e[16];
  float s = 0.f;
  #pragma unroll
  for (int j = 0; j < 16; ++j) { e[j] = __expf(p[j] - m); s += e[j]; }
  const float inv = 1.f / s;
  #pragma unroll
  for (int j = 0; j < 16; ++j) p[j] = e[j] * inv;
}

// ---------------------------------------------------------------------------
// Deformable sampling: one wave per (query row, head); lane = head-dim channel.
// V layout: [vrow][head*32 + hd] -> each corner gather is a coalesced 128B read.
// ---------------------------------------------------------------------------
__global__ __launch_bounds__(256)
void sample_kernel(const float* __restrict__ off, const float* __restrict__ att,
                   const float* __restrict__ V, float* __restrict__ agg)
{
  const int head = threadIdx.x >> 5;
  const int lane = threadIdx.x & 31;
  const int row  = blockIdx.x;

  int lq, bt, n;
  row_to_lbn(row, lq, bt, n);

  constexpr int   DIM[4]  = {64, 32, 16, 8};
  constexpr int   LOGW[4] = {6, 5, 4, 3};
  constexpr int   RB[4]   = {0, 16384, 20480, 21504};   // V row bases (BT*cumHW)
  constexpr int   HW[4]   = {4096, 1024, 256, 64};
  constexpr float INVD[4] = {1.f/64.f, 1.f/32.f, 1.f/16.f, 1.f/8.f};

  const int   Wq   = DIM[lq];
  const int   qy   = n >> LOGW[lq];
  const int   qx   = n & (Wq - 1);
  const float refx = ((float)qx + 0.5f) * INVD[lq];
  const float refy = ((float)qy + 0.5f) * INVD[lq];

  const float* offp = off + (size_t)row * EMBED + head * HEAD_DIM;
  const float* awp  = att + (size_t)row * (NUM_HEADS * 16) + head * 16;
  const int    vcol = head * HEAD_DIM + lane;

  float acc = 0.f;
  #pragma unroll
  for (int l = 0; l < NUM_LEVELS; ++l) {
    const int    Wl    = DIM[l];
    const int    Hl    = DIM[l];
    const float  fD    = (float)DIM[l];
    const float  invD  = INVD[l];
    const size_t vrow0 = (size_t)(RB[l] + bt * HW[l]);
    #pragma unroll
    for (int k = 0; k < NUM_POINTS; ++k) {
      const float ox = offp[l * 8 + k * 2 + 0];
      const float oy = offp[l * 8 + k * 2 + 1];
      const float aw = awp[l * 4 + k];
      float lx = fminf(fmaxf(refx + ox * invD, 0.f), 1.f);
      float ly = fminf(fmaxf(refy + oy * invD, 0.f), 1.f);
      const float x  = lx * fD - 0.5f;           // align_corners=False pixel coord
      const float y  = ly * fD - 0.5f;
      const float xf = floorf(x), yf = floorf(y);
      const int   x0 = (int)xf, y0 = (int)yf;
      const float fx = x - xf,  fy = y - yf;
      const float w00 = (1.f - fx) * (1.f - fy);
      const float w10 = fx * (1.f - fy);
      const float w01 = (1.f - fx) * fy;
      const float w11 = fx * fy;
      const bool vx0 = (x0 >= 0) & (x0 < Wl);
      const bool vx1 = (x0 + 1 >= 0) & (x0 + 1 < Wl);
      const bool vy0 = (y0 >= 0) & (y0 < Hl);
      const bool vy1 = (y0 + 1 >= 0) & (y0 + 1 < Hl);
      if (vx0 & vy0) acc += aw * w00 * V[(vrow0 + (size_t)(y0 * Wl + x0)) * EMBED + vcol];
      if (vx1 & vy0) acc += aw * w10 * V[(vrow0 + (size_t)(y0 * Wl + x0 + 1)) * EMBED + vcol];
      if (vx0 & vy1) acc += aw * w01 * V[(vrow0 + (size_t)((y0 + 1) * Wl + x0)) * EMBED + vcol];
      if (vx1 & vy1) acc += aw * w11 * V[(vrow0 + (size_t)((y0 + 1) * Wl + x0 + 1)) * EMBED + vcol];
    }
  }
  agg[(size_t)row * EMBED + vcol] = acc;
}

// ---------------------------------------------------------------------------
extern "C" void kernel_launch(void* const* d_in, const int* in_sizes, int n_in,
                              void* d_out, int out_size, void* d_ws, size_t ws_size,
                              hipStream_t stream)
{
  const float* f0    = (const float*)d_in[0];
  const float* f1    = (const float*)d_in[1];
  const float* f2    = (const float*)d_in[2];
  const float* f3    = (const float*)d_in[3];
  const float* Wq    = (const float*)d_in[4];
  const float* bq    = (const float*)d_in[5];
  const float* Wv    = (const float*)d_in[6];
  const float* bv    = (const float*)d_in[7];
  const float* Woff  = (const float*)d_in[8];
  const float* boff  = (const float*)d_in[9];
  const float* Wattn = (const float*)d_in[10];
  const float* battn = (const float*)d_in[11];
  const float* Wout  = (const float*)d_in[12];
  const float* bout  = (const float*)d_in[13];
  float*       out   = (float*)d_out;

  // workspace layout (floats): V | Q | OFF | ATT | AGG
  float* V   = (float*)d_ws;
  float* Q   = V   + (size_t)ROWS_TOTAL * EMBED;
  float* OFF = Q   + (size_t)ROWS_TOTAL * EMBED;
  float* ATT = OFF + (size_t)ROWS_TOTAL * EMBED;
  float* AGG = ATT + (size_t)ROWS_TOTAL * (NUM_HEADS * 16);

  const dim3 blk(256);
  const dim3 g256(ROW_BLKS, 2);   // 256 output cols
  const dim3 g128(ROW_BLKS, 1);   // 128 output cols

  // 1-2: value + query projections (A streamed from feats)
  gemm_wmma_kernel<256, 1, 0><<<g256, blk, 0, stream>>>(nullptr, f0, f1, f2, f3, Wv, bv, V);
  gemm_wmma_kernel<256, 1, 0><<<g256, blk, 0, stream>>>(nullptr, f0, f1, f2, f3, Wq, bq, Q);
  // 3-4: offset + attention projections (A tile via TDM)
  gemm_wmma_kernel<256, 0, 0><<<g256, blk, 0, stream>>>(Q, nullptr, nullptr, nullptr, nullptr, Woff, boff, OFF);
  gemm_wmma_kernel<128, 0, 0><<<g128, blk, 0, stream>>>(Q, nullptr, nullptr, nullptr, nullptr, Wattn, battn, ATT);
  // 5: softmax over 16 logits per (row, head)
  {
    const int total = ROWS_TOTAL * NUM_HEADS;
    softmax16_kernel<<<(total + 255) / 256, blk, 0, stream>>>(ATT, total);
  }
  // 6: deformable bilinear sampling + weighted aggregation
  sample_kernel<<<ROWS_TOTAL, blk, 0, stream>>>(OFF, ATT, V, AGG);
  // 7: output projection, scattered directly into [B,C,H,W,T] outputs
  gemm_wmma_kernel<256, 0, 1><<<g256, blk, 0, stream>>>(AGG, nullptr, nullptr, nullptr, nullptr, Wout, bout, out);
}